// CrossScanStrategy_7662221656541
// MI455X (gfx1250) — compile-verified
//
#include <hip/hip_runtime.h>
#include <math.h>

typedef __attribute__((ext_vector_type(2))) float v2f;
typedef __attribute__((ext_vector_type(8))) float v8f;

#define BB 2
#define CC 256
#define HH 64
#define WW 64
#define HW 4096
#define NN 16
#define NROW (BB*HW)   // 8192

// ---------------- out = features * D (one-time x*D term; zero-bases atomics) --
__global__ void init_out_kernel(const float* __restrict__ feat,
                                const float* __restrict__ Dvec,
                                float* __restrict__ out) {
    int idx = blockIdx.x * blockDim.x + threadIdx.x;   // over B*C*HW = 2^21
    int c = (idx >> 12) & (CC - 1);                    // layout (B, C, HW)
    out[idx] = feat[idx] * Dvec[c];
}

// ---------------- projections via V_WMMA_F32_16X16X4_F32 ---------------------
// One wave = one 16x16 tile of delta_pre (or one 16x16 Bm/Cm tile).
// x[row][k] = feat[b*C*HW + k*HW + l] with row = b*HW + l (row-major (BL, C) view).
// A fragment (16x4 f32, ISA 7.12.2): lane m=lane&15, VGPR{0,1} hold K = 2*(lane>>4)+{0,1}
// C/D fragment: lane n=lane&15, VGPR v holds M = v + 8*(lane>>4)
__global__ void proj_wmma_kernel(const float* __restrict__ feat,
                                 const float* __restrict__ W_delta,
                                 const float* __restrict__ b_delta,
                                 const float* __restrict__ W_B,
                                 const float* __restrict__ W_C,
                                 float* __restrict__ delta_ws,
                                 float* __restrict__ Bm_ws,
                                 float* __restrict__ Cm_ws) {
    const int lane = threadIdx.x;      // 0..31
    const int lo   = lane & 15;
    const int hi   = lane >> 4;        // 0/1
    const int m0   = blockIdx.x * 16;  // row tile base
    const int which = blockIdx.y;      // 0..15: delta col tiles; 16: Bm; 17: Cm

    const float* Wmat;
    int ncols, n0;
    if (which < 16)       { Wmat = W_delta; ncols = CC; n0 = which * 16; }
    else if (which == 16) { Wmat = W_B;     ncols = NN; n0 = 0; }
    else                  { Wmat = W_C;     ncols = NN; n0 = 0; }

    const int rowA = m0 + lo;              // global (b*HW + l) row for A
    const int bA   = rowA >> 12;
    const int lA   = rowA & (HW - 1);
    const float* aBase = feat + (size_t)bA * CC * HW + lA;   // + k*HW strides

    v8f acc = {};
    for (int k0 = 0; k0 < CC; k0 += 4) {
        const int ka = k0 + hi * 2;
        v2f a, b;
        a.x = aBase[(size_t)ka * HW];
        a.y = aBase[(size_t)(ka + 1) * HW];
        b.x = Wmat[(size_t)ka * ncols + n0 + lo];
        b.y = Wmat[(size_t)(ka + 1) * ncols + n0 + lo];
        acc = __builtin_amdgcn_wmma_f32_16x16x4_f32(
            /*neg_a=*/false, a, /*neg_b=*/false, b,
            /*c_mod=*/(short)0, acc, /*reuse_a=*/false, /*reuse_b=*/false);
    }

    if (which < 16) {
        const float bias = b_delta[n0 + lo];
        #pragma unroll
        for (int v = 0; v < 8; ++v) {
            const int row = m0 + v + 8 * hi;
            const float pre = acc[v] + bias;
            const float sp = (pre > 20.f) ? pre : log1pf(__expf(pre)); // softplus
            delta_ws[(size_t)row * CC + n0 + lo] = sp;
        }
    } else {
        float* dst = (which == 16) ? Bm_ws : Cm_ws;
        #pragma unroll
        for (int v = 0; v < 8; ++v) {
            const int row = m0 + v + 8 * hi;
            dst[(size_t)row * NN + lo] = acc[v];
        }
    }
}

// ---------------- 4-direction selective scan ---------------------------------
// One wave handles two (b,c) chains: lanes [0,16) -> c0, [16,32) -> c0+1;
// lane&15 = state index n. Per step, reduce h*Cm over n and atomically add /4.
__global__ void scan_kernel(const float* __restrict__ feat,
                            const float* __restrict__ A_log,
                            const float* __restrict__ delta_ws,
                            const float* __restrict__ Bm_ws,
                            const float* __restrict__ Cm_ws,
                            float* __restrict__ out) {
    const int lane = threadIdx.x;
    const int n  = lane & 15;
    const int hi = lane >> 4;
    const int id  = blockIdx.x;          // 0..1023
    const int dir = id >> 8;             // 0..3
    const int rem = id & 255;
    const int b   = rem >> 7;            // 0..1
    const int c   = ((rem & 127) << 1) + hi;   // 0..255

    // direction index map: l = base + sh*(t>>6) + sw*(t&63)
    int base, sh, sw;
    switch (dir) {
        case 0:  base = 0;    sh =  64; sw =  1; break;  // row-major forward
        case 1:  base = 4095; sh = -64; sw = -1; break;  // reversed
        case 2:  base = 4032; sh = -64; sw =  1; break;  // H flipped
        default: base = 63;   sh =  -1; sw = 64; break;  // col-major, W flipped
    }

    const float A = -__expf(A_log[c * NN + n]);
    const float* fBase = feat + (size_t)b * CC * HW + (size_t)c * HW;
    const size_t rbase = (size_t)b * HW;
    float* outBase = out + (size_t)b * CC * HW + (size_t)c * HW;

    float h = 0.f;
    for (int t = 0; t < HW; ++t) {
        const int l = base + sh * (t >> 6) + sw * (t & 63);
        const size_t row = rbase + (size_t)l;
        const float d  = delta_ws[row * CC + c];   // broadcast across 16 lanes
        const float xv = fBase[l];                 // broadcast across 16 lanes
        const float Bv = Bm_ws[row * NN + n];
        const float Cv = Cm_ws[row * NN + n];
        const float dA = __expf(d * A);
        h = fmaf(dA, h, d * xv * Bv);
        float contrib = h * Cv;
        contrib += __shfl_xor(contrib, 1);
        contrib += __shfl_xor(contrib, 2);
        contrib += __shfl_xor(contrib, 4);
        contrib += __shfl_xor(contrib, 8);         // sum over n within 16-lane half
        if (n == 0) atomicAdd(&outBase[l], contrib * 0.25f);
    }
}

extern "C" void kernel_launch(void* const* d_in, const int* in_sizes, int n_in,
                              void* d_out, int out_size, void* d_ws, size_t ws_size,
                              hipStream_t stream) {
    const float* feat    = (const float*)d_in[0];  // (B, C, H, W)
    const float* A_log   = (const float*)d_in[1];  // (C, N)
    const float* Dvec    = (const float*)d_in[2];  // (C,)
    const float* W_delta = (const float*)d_in[3];  // (C, C)
    const float* b_delta = (const float*)d_in[4];  // (C,)
    const float* W_B     = (const float*)d_in[5];  // (C, N)
    const float* W_C     = (const float*)d_in[6];  // (C, N)
    float* out = (float*)d_out;

    float* ws       = (float*)d_ws;
    float* delta_ws = ws;                                   // 8192*256 floats
    float* Bm_ws    = ws + (size_t)NROW * CC;               // 8192*16
    float* Cm_ws    = Bm_ws + (size_t)NROW * NN;            // 8192*16

    init_out_kernel<<<(BB * CC * HW) / 256, 256, 0, stream>>>(feat, Dvec, out);
    proj_wmma_kernel<<<dim3(NROW / 16, 18), 32, 0, stream>>>(
        feat, W_delta, b_delta, W_B, W_C, delta_ws, Bm_ws, Cm_ws);
    scan_kernel<<<1024, 32, 0, stream>>>(feat, A_log, delta_ws, Bm_ws, Cm_ws, out);
}